// xLSTM_Block_62654982914536
// MI455X (gfx1250) — compile-verified
//
#include <hip/hip_runtime.h>
#include <math.h>

// ---------------------------------------------------------------------------
// xLSTM 2-layer mLSTM block, fused for MI455X (gfx1250).
// All GEMVs run on V_WMMA_F32_16X16X4_F32 (fp32 matrix pipe, wave32).
// Algebraic fusion avoids materializing c1:
//   c1@q = f1*(c0@q) + i1*(k1.q)*v1
//   c2   = f1*f2*c0 + f2*i1*(v1 k1^T) + i2*(v2 k2^T)
// Deep split-K (8 segments) keeps ~8x more loads in flight for the
// bandwidth-bound weight streams (23.3 TB/s x ~800ns => ~18MB needed).
// ---------------------------------------------------------------------------

typedef __attribute__((ext_vector_type(2))) float v2f;
typedef __attribute__((ext_vector_type(8))) float v8f;

#define N_DIM 4096
#define SEGS 8                 // split-K segments per GEMV
#define SEG_LEN (N_DIM / SEGS) // 512
#define INV_SQRT_M 0.015625f   // 1/sqrt(4096)

struct Reduce4Args {
    const float* bias[4];
    float*       out[4];
    float        scale[4];
};

__device__ inline float sigf(float x) { return 1.0f / (1.0f + expf(-x)); }

// One wave computes partial dot products for rows [row0, row0+16) over
// K range [kbase, kbase+klen) using V_WMMA_F32_16X16X4_F32.
// A fragment (16x4 f32): lane L<16 -> (M=L, K={0,1}); lane L>=16 -> (M=L-16, K={2,3}).
// B fragment (4x16 f32): B[k][n] = x[kbase+k] for all n (columns identical).
// D (16x16 f32): vgpr j, lanes 0-15 -> M=j; lanes 16-31 -> M=j+8. Column N=0 read out.
__device__ inline void gemv16_wmma(const float* __restrict__ W,
                                   const float* __restrict__ x,
                                   int row0, int kbase, int klen, int ld,
                                   float* __restrict__ out) {
    const int lane = threadIdx.x & 31;
    const int hi   = lane >> 4;      // 0: K pair {0,1}; 1: K pair {2,3}
    const int mrow = lane & 15;

    const float* wp = W + (size_t)(row0 + mrow) * (size_t)ld + kbase + 2 * hi;
    const float* xp = x + kbase + 2 * hi;

    v8f a0 = {}, a1 = {}, a2 = {}, a3 = {};
    for (int k = 0; k < klen; k += 16) {
        __builtin_prefetch(wp + k + 1024, 0, 0);   // ~4KB ahead in this row
        v2f wA0 = *(const v2f*)(wp + k);
        v2f xB0 = *(const v2f*)(xp + k);
        v2f wA1 = *(const v2f*)(wp + k + 4);
        v2f xB1 = *(const v2f*)(xp + k + 4);
        v2f wA2 = *(const v2f*)(wp + k + 8);
        v2f xB2 = *(const v2f*)(xp + k + 8);
        v2f wA3 = *(const v2f*)(wp + k + 12);
        v2f xB3 = *(const v2f*)(xp + k + 12);
        a0 = __builtin_amdgcn_wmma_f32_16x16x4_f32(false, wA0, false, xB0, (short)0, a0, false, false);
        a1 = __builtin_amdgcn_wmma_f32_16x16x4_f32(false, wA1, false, xB1, (short)0, a1, false, false);
        a2 = __builtin_amdgcn_wmma_f32_16x16x4_f32(false, wA2, false, xB2, (short)0, a2, false, false);
        a3 = __builtin_amdgcn_wmma_f32_16x16x4_f32(false, wA3, false, xB3, (short)0, a3, false, false);
    }
    v8f acc = (a0 + a1) + (a2 + a3);

    if (mrow == 0) {                 // lane 0 -> rows row0..+7 ; lane 16 -> rows +8..+15
        const int mb = row0 + (hi ? 8 : 0);
        #pragma unroll
        for (int j = 0; j < 8; ++j) out[mb + j] = acc[j];
    }
}

// 4 stacked GEMVs (q,k,v,o) over the same x, split-K by SEGS.
// grid = 4*256*SEGS waves = 2048 blocks of 128 (4 waves).
// Partials: part[((mat*SEGS + seg) * N_DIM) + row]
__global__ void __launch_bounds__(128) gemv4_splitk_kernel(const float* __restrict__ W0,
                                                           const float* __restrict__ W1,
                                                           const float* __restrict__ W2,
                                                           const float* __restrict__ W3,
                                                           const float* __restrict__ x,
                                                           float* __restrict__ part) {
    const int wave = threadIdx.x >> 5;
    const int gid  = blockIdx.x * 4 + wave;     // [0, 8192)
    const int mat  = gid >> 11;                 // 2048 waves per matrix
    const int rem  = gid & 2047;
    const int seg  = rem & (SEGS - 1);
    const int rg   = rem >> 3;                  // [0, 256)
    const float* W = (mat == 0) ? W0 : (mat == 1) ? W1 : (mat == 2) ? W2 : W3;
    gemv16_wmma(W, x, rg * 16, seg * SEG_LEN, SEG_LEN, N_DIM,
                part + (size_t)((mat << 3) + seg) * N_DIM);
}

// Reduce split-K partials for the 4-matrix GEMV: out = scale*(sum + bias).
__global__ void __launch_bounds__(256) reduce4_kernel(const float* __restrict__ part,
                                                      Reduce4Args args) {
    const int t   = blockIdx.x * blockDim.x + threadIdx.x;  // [0, 16384)
    const int mat = t >> 12;
    const int row = t & (N_DIM - 1);
    float s = 0.0f;
    #pragma unroll
    for (int g = 0; g < SEGS; ++g)
        s += part[(size_t)((mat << 3) + g) * N_DIM + row];
    args.out[mat][row] = args.scale[mat] * (s + args.bias[mat][row]);
}

// Split-K GEMV over c0 (4096x4096): SEGS K-segments, partials to ws.
// grid = 256*SEGS waves = 512 blocks of 128.
__global__ void __launch_bounds__(128) gemv_splitk_kernel(const float* __restrict__ C,
                                                          const float* __restrict__ q,
                                                          float* __restrict__ part) {
    const int wave = threadIdx.x >> 5;
    const int gid  = blockIdx.x * 4 + wave;     // [0, 2048)
    const int seg  = gid & (SEGS - 1);
    const int rg   = gid >> 3;
    gemv16_wmma(C, q, rg * 16, seg * SEG_LEN, SEG_LEN, N_DIM,
                part + (size_t)seg * N_DIM);
}

__global__ void reduce_part(const float* __restrict__ part, float* __restrict__ u) {
    const int t = blockIdx.x * blockDim.x + threadIdx.x;
    float s = 0.0f;
    #pragma unroll
    for (int g = 0; g < SEGS; ++g) s += part[(size_t)g * N_DIM + t];
    u[t] = s;
}

// Single-block dot product (+ optional scalar bias), length n, result -> slot.
__global__ void __launch_bounds__(256) dot_kernel(const float* __restrict__ a,
                                                  const float* __restrict__ b,
                                                  const float* __restrict__ bias,
                                                  float* __restrict__ slot, int n) {
    __shared__ float red[256];
    float s = 0.0f;
    for (int t = threadIdx.x; t < n; t += 256) s += a[t] * b[t];
    red[threadIdx.x] = s;
    __syncthreads();
    for (int o = 128; o > 0; o >>= 1) {
        if ((int)threadIdx.x < o) red[threadIdx.x] += red[threadIdx.x + o];
        __syncthreads();
    }
    if (threadIdx.x == 0) slot[0] = red[0] + (bias ? bias[0] : 0.0f);
}

// scal slots: 0=gi1 1=gf1 2=n0.q1 3=k1.q1 4=gi2 5=gf2 6=k1.q2 7=k2.q2 8=n1.q2
__global__ void finish1(const float* __restrict__ scal, const float* __restrict__ go1,
                        const float* __restrict__ u1, const float* __restrict__ v1,
                        const float* __restrict__ k1, const float* __restrict__ n0,
                        float* __restrict__ h1, float* __restrict__ n1) {
    const int t = blockIdx.x * blockDim.x + threadIdx.x;
    const float i1 = expf(scal[0]);
    const float f1 = sigf(scal[1]);
    const float kq = scal[3];
    const float denom = fmaxf(fabsf(f1 * scal[2] + i1 * kq), 1.0f);
    const float ht = (f1 * u1[t] + i1 * kq * v1[t]) / denom;
    h1[t] = sigf(go1[t]) * ht;
    n1[t] = f1 * n0[t] + i1 * k1[t];
}

__global__ void finish2(const float* __restrict__ scal, const float* __restrict__ go2,
                        const float* __restrict__ u2, const float* __restrict__ v1,
                        const float* __restrict__ v2, const float* __restrict__ k2,
                        const float* __restrict__ n1, float* __restrict__ outH,
                        float* __restrict__ outN) {
    const int t = blockIdx.x * blockDim.x + threadIdx.x;
    const float i1 = expf(scal[0]);
    const float f1 = sigf(scal[1]);
    const float i2 = expf(scal[4]);
    const float f2 = sigf(scal[5]);
    const float k1q2 = scal[6], k2q2 = scal[7], n1q2 = scal[8];
    const float denom = fmaxf(fabsf(f2 * n1q2 + i2 * k2q2), 1.0f);
    const float ctq = f2 * f1 * u2[t] + f2 * i1 * k1q2 * v1[t] + i2 * k2q2 * v2[t];
    outH[t] = sigf(go2[t]) * ctq / denom;
    outN[t] = f2 * n1[t] + i2 * k2[t];
}

// c2 = (f1*f2)*c0 + (f2*i1)*(v1 k1^T) + i2*(v2 k2^T)   (float4-vectorized stream)
__global__ void __launch_bounds__(256) c2_kernel(const float* __restrict__ scal,
                                                 const float* __restrict__ c0,
                                                 const float* __restrict__ v1,
                                                 const float* __restrict__ k1,
                                                 const float* __restrict__ v2,
                                                 const float* __restrict__ k2,
                                                 float* __restrict__ c2) {
    const size_t idx = (size_t)blockIdx.x * blockDim.x + threadIdx.x; // float4 index
    const int m  = (int)(idx >> 10);          // 1024 float4 per 4096-row
    const int j  = (int)(idx & 1023) << 2;
    const float i1 = expf(scal[0]);
    const float f1 = sigf(scal[1]);
    const float i2 = expf(scal[4]);
    const float f2 = sigf(scal[5]);
    const float A  = f1 * f2;
    const float B1 = f2 * i1 * v1[m];
    const float B2 = i2 * v2[m];
    const float4 cc = *(const float4*)(c0 + (size_t)m * N_DIM + j);
    const float4 a  = *(const float4*)(k1 + j);
    const float4 b  = *(const float4*)(k2 + j);
    float4 r;
    r.x = A * cc.x + B1 * a.x + B2 * b.x;
    r.y = A * cc.y + B1 * a.y + B2 * b.y;
    r.z = A * cc.z + B1 * a.z + B2 * b.z;
    r.w = A * cc.w + B1 * a.w + B2 * b.w;
    *(float4*)(c2 + (size_t)m * N_DIM + j) = r;
}

// ---------------------------------------------------------------------------

extern "C" void kernel_launch(void* const* d_in, const int* in_sizes, int n_in,
                              void* d_out, int out_size, void* d_ws, size_t ws_size,
                              hipStream_t stream) {
    (void)in_sizes; (void)n_in; (void)out_size; (void)ws_size;

    const float* x  = (const float*)d_in[0];
    const float* c0 = (const float*)d_in[1];
    const float* n0 = (const float*)d_in[2];
    // layer 0
    const float* w_q0 = (const float*)d_in[3];   const float* b_q0 = (const float*)d_in[4];
    const float* w_k0 = (const float*)d_in[5];   const float* b_k0 = (const float*)d_in[6];
    const float* w_v0 = (const float*)d_in[7];   const float* b_v0 = (const float*)d_in[8];
    const float* w_i0 = (const float*)d_in[9];   const float* b_i0 = (const float*)d_in[10];
    const float* w_f0 = (const float*)d_in[11];  const float* b_f0 = (const float*)d_in[12];
    const float* w_o0 = (const float*)d_in[13];  const float* b_o0 = (const float*)d_in[14];
    // layer 1
    const float* w_q1 = (const float*)d_in[15];  const float* b_q1 = (const float*)d_in[16];
    const float* w_k1 = (const float*)d_in[17];  const float* b_k1 = (const float*)d_in[18];
    const float* w_v1 = (const float*)d_in[19];  const float* b_v1 = (const float*)d_in[20];
    const float* w_i1 = (const float*)d_in[21];  const float* b_i1 = (const float*)d_in[22];
    const float* w_f1 = (const float*)d_in[23];  const float* b_f1 = (const float*)d_in[24];
    const float* w_o1 = (const float*)d_in[25];  const float* b_o1 = (const float*)d_in[26];

    float* ws = (float*)d_ws;
    float* q1   = ws + 0 * N_DIM;
    float* k1   = ws + 1 * N_DIM;
    float* v1   = ws + 2 * N_DIM;
    float* go1  = ws + 3 * N_DIM;
    float* h1   = ws + 4 * N_DIM;
    float* q2   = ws + 5 * N_DIM;
    float* k2   = ws + 6 * N_DIM;
    float* v2   = ws + 7 * N_DIM;
    float* go2  = ws + 8 * N_DIM;
    float* u1   = ws + 9 * N_DIM;
    float* u2   = ws + 10 * N_DIM;
    float* n1   = ws + 11 * N_DIM;
    float* part = ws + 12 * N_DIM;              // up to 4*SEGS*N_DIM = 131072 floats
    float* scal = ws + 12 * N_DIM + 4 * SEGS * N_DIM;   // 9 scalar slots

    float* outH = (float*)d_out;
    float* outC = outH + N_DIM;
    float* outN = outC + (size_t)N_DIM * N_DIM;

    // ---- Layer 0 projections: q1, k1, v1, o1(pre) ----
    gemv4_splitk_kernel<<<2048, 128, 0, stream>>>(w_q0, w_k0, w_v0, w_o0, x, part);
    Reduce4Args r0;
    r0.bias[0] = b_q0; r0.out[0] = q1;  r0.scale[0] = 1.0f;
    r0.bias[1] = b_k0; r0.out[1] = k1;  r0.scale[1] = INV_SQRT_M;
    r0.bias[2] = b_v0; r0.out[2] = v1;  r0.scale[2] = 1.0f;
    r0.bias[3] = b_o0; r0.out[3] = go1; r0.scale[3] = 1.0f;
    reduce4_kernel<<<64, 256, 0, stream>>>(part, r0);
    dot_kernel<<<1, 256, 0, stream>>>(w_i0, x, b_i0, scal + 0, N_DIM);
    dot_kernel<<<1, 256, 0, stream>>>(w_f0, x, b_f0, scal + 1, N_DIM);

    // ---- u1 = c0 @ q1 (split-K WMMA GEMV + deterministic reduce) ----
    gemv_splitk_kernel<<<512, 128, 0, stream>>>(c0, q1, part);
    reduce_part<<<16, 256, 0, stream>>>(part, u1);
    dot_kernel<<<1, 256, 0, stream>>>(n0, q1, nullptr, scal + 2, N_DIM);
    dot_kernel<<<1, 256, 0, stream>>>(k1, q1, nullptr, scal + 3, N_DIM);

    // ---- h1, n1 ----
    finish1<<<16, 256, 0, stream>>>(scal, go1, u1, v1, k1, n0, h1, n1);

    // ---- Layer 1 projections on h1 ----
    gemv4_splitk_kernel<<<2048, 128, 0, stream>>>(w_q1, w_k1, w_v1, w_o1, h1, part);
    Reduce4Args r1;
    r1.bias[0] = b_q1; r1.out[0] = q2;  r1.scale[0] = 1.0f;
    r1.bias[1] = b_k1; r1.out[1] = k2;  r1.scale[1] = INV_SQRT_M;
    r1.bias[2] = b_v1; r1.out[2] = v2;  r1.scale[2] = 1.0f;
    r1.bias[3] = b_o1; r1.out[3] = go2; r1.scale[3] = 1.0f;
    reduce4_kernel<<<64, 256, 0, stream>>>(part, r1);
    dot_kernel<<<1, 256, 0, stream>>>(w_i1, h1, b_i1, scal + 4, N_DIM);
    dot_kernel<<<1, 256, 0, stream>>>(w_f1, h1, b_f1, scal + 5, N_DIM);

    // ---- u2 = c0 @ q2 ----
    gemv_splitk_kernel<<<512, 128, 0, stream>>>(c0, q2, part);
    reduce_part<<<16, 256, 0, stream>>>(part, u2);
    dot_kernel<<<1, 256, 0, stream>>>(k1, q2, nullptr, scal + 6, N_DIM);
    dot_kernel<<<1, 256, 0, stream>>>(k2, q2, nullptr, scal + 7, N_DIM);
    dot_kernel<<<1, 256, 0, stream>>>(n1, q2, nullptr, scal + 8, N_DIM);

    // ---- h2 (out), n2 (out) ----
    finish2<<<16, 256, 0, stream>>>(scal, go2, u2, v1, v2, k2, n1, outH, outN);

    // ---- c2 (out): scaled c0 + rank-2 outer update, 128 MB stream ----
    c2_kernel<<<16384, 256, 0, stream>>>(scal, c0, v1, k1, v2, k2, outC);
}